// ConvAttention_22574348108555
// MI455X (gfx1250) — compile-verified
//
#include <hip/hip_runtime.h>

// ConvAttention for MI455X (gfx1250): all GEMMs via V_WMMA_F32_16X16X4_F32.
// Stage 0: repack conv/proj weights into WMMA B-fragment order (coalesced b64).
// Stage 1: fused q/v/k convs (1x1, 3x3, 5x5) as implicit GEMMs, 2x2 register
//          blocking (2 pixel-tiles x 2 outchannel-tiles per wave -> 4 wmma per
//          4 b64 loads), output stored directly in the reference's (quirky)
//          attention layout [b,head,m,d].
// Stage 2: flash-attention per (b,head), online softmax, LDS-staged fragments.
// Stage 3: 256x256 output projection GEMM.

typedef __attribute__((ext_vector_type(2))) float v2f;
typedef __attribute__((ext_vector_type(8))) float v8f;

static __device__ __forceinline__ v8f wmma_f32(v2f a, v2f b, v8f c) {
  // 8 args: (neg_a, A, neg_b, B, c_mod, C, reuse_a, reuse_b)
  return __builtin_amdgcn_wmma_f32_16x16x4_f32(false, a, false, b, (short)0, c,
                                               false, false);
}

// ---------------------------------------------------------------------------
// Stage 0: weight repack.
// B-fragment layout (16x16x4 f32): lane l, vgpr j holds B[K=2*j+(l>>4)][N=l&15].
// Packed layout: wpk[(((tap*64 + kchunk)*16 + otile)*32 + lane)*2 + j]
//   = w[o = otile*16 + (lane&15)][i = kchunk*4 + 2*j + (lane>>4)][tap]
// so the conv/proj inner loop reads one coalesced b64 per fragment.
// ---------------------------------------------------------------------------
__global__ __launch_bounds__(256) void pack_w_kernel(
    const float* __restrict__ w, float* __restrict__ out, int KK, int total) {
  int e = blockIdx.x * 256 + threadIdx.x;
  if (e >= total) return;
  int j = e & 1;
  int lane = (e >> 1) & 31;
  int otile = (e >> 6) & 15;
  int kchunk = (e >> 10) & 63;
  int tap = e >> 16;
  int o = otile * 16 + (lane & 15);
  int i = kchunk * 4 + 2 * j + (lane >> 4);
  out[e] = w[(size_t)(o * 256 + i) * KK + tap];
}

// Reference reshape: conv out (b, c, n) -> attention tensor [b, head, m, d]
//   head = c>>4 ; m = (c&15)*64 + (n>>4) ; d = n&15
static __device__ __forceinline__ void qkv_store(float* __restrict__ buf,
                                                 int b, int o, int n, float v) {
  int head = o >> 4;
  int m = (o & 15) * 64 + (n >> 4);
  int d = n & 15;
  buf[(((size_t)b * 16 + head) * 1024 + m) * 16 + d] = v;
}

// One image row (32 pixels = 2 M-tiles) x 32 out-channels (2 N-tiles) per wave.
template <int KS>
static __device__ void conv_row(const float* __restrict__ x,
                                const float* __restrict__ wpk,  // packed
                                const float* __restrict__ bia,
                                float* __restrict__ dst,
                                int b, int h, int lane) {
  const int lrow = lane & 15;      // A: M row / B: N col
  const int lhalf = lane >> 4;     // half-wave selector
  const int wave = threadIdx.x >> 5;
  const int ot0 = wave * 2;        // two 16-wide out-channel tiles per wave
  const int o0 = ot0 * 16;
  const int R = KS / 2;

  v8f acc00 = {};  // [pixel-tile 0][o-tile 0]
  v8f acc01 = {};  // [pixel-tile 0][o-tile 1]
  v8f acc10 = {};  // [pixel-tile 1][o-tile 0]
  v8f acc11 = {};  // [pixel-tile 1][o-tile 1]

  int tap = 0;
  for (int ky = 0; ky < KS; ++ky) {
    int hh = h + ky - R;
    bool hin = (hh >= 0) && (hh < 32);
    for (int kx = 0; kx < KS; ++kx, ++tap) {
      int ww0 = kx - R + lrow;        // pixel tile 0: w in [0,16)
      int ww1 = ww0 + 16;             // pixel tile 1: w in [16,32)
      bool in0 = hin && (ww0 >= 0) && (ww0 < 32);   // SAME padding
      bool in1 = hin && (ww1 >= 0) && (ww1 < 32);
      const float* xrow =
          x + ((size_t)b * 32 + (hin ? hh : 0)) * 32 * 256 + 2 * lhalf;
      const float* xp0 = xrow + (size_t)(in0 ? ww0 : 0) * 256;
      const float* xp1 = xrow + (size_t)(in1 ? ww1 : 0) * 256;
      // packed weights for this tap; per-kchunk stride = 16 tiles * 64 floats
      const float* wt = wpk + (size_t)tap * 64 * 16 * 64;
      const float* wt0 = wt + ((size_t)ot0 * 32 + lane) * 2;
      const float* wt1 = wt0 + 64;  // next o-tile
#pragma unroll 4
      for (int c = 0; c < 64; ++c) {
        // A fragments: lane l, vgpr j -> x[pixel = lrow][chan = 4c+2*lhalf+j]
        v2f a0 = {0.0f, 0.0f};
        v2f a1 = {0.0f, 0.0f};
        if (in0) a0 = *(const v2f*)(xp0 + c * 4);
        if (in1) a1 = *(const v2f*)(xp1 + c * 4);
        // B fragments: one coalesced b64 each from packed weights
        v2f b0 = *(const v2f*)(wt0 + (size_t)c * 1024);
        v2f b1 = *(const v2f*)(wt1 + (size_t)c * 1024);
        acc00 = wmma_f32(a0, b0, acc00);
        acc01 = wmma_f32(a0, b1, acc01);
        acc10 = wmma_f32(a1, b0, acc10);
        acc11 = wmma_f32(a1, b1, acc11);
      }
    }
  }

  float bi0 = bia[o0 + lrow];
  float bi1 = bia[o0 + 16 + lrow];
#pragma unroll
  for (int r = 0; r < 8; ++r) {
    int n0 = h * 32 + r + 8 * lhalf;       // D row = pixel within tile
    int n1 = n0 + 16;
    qkv_store(dst, b, o0 + lrow,      n0, acc00[r] + bi0);
    qkv_store(dst, b, o0 + 16 + lrow, n0, acc01[r] + bi1);
    qkv_store(dst, b, o0 + lrow,      n1, acc10[r] + bi0);
    qkv_store(dst, b, o0 + 16 + lrow, n1, acc11[r] + bi1);
  }
}

__global__ __launch_bounds__(256) void conv_qkv_kernel(
    const float* __restrict__ x,
    const float* __restrict__ wpk3, const float* __restrict__ b3,
    const float* __restrict__ wpk5, const float* __restrict__ b5,
    const float* __restrict__ wpk7, const float* __restrict__ b7,
    float* __restrict__ Qa, float* __restrict__ Va, float* __restrict__ Ka) {
  int t = blockIdx.x;              // 256 row-jobs: one image row (b, h)
  int b = t >> 5;
  int h = t & 31;
  int lane = threadIdx.x & 31;
  conv_row<1>(x, wpk3, b3, Qa, b, h, lane);  // q = 1x1 conv
  conv_row<3>(x, wpk5, b5, Va, b, h, lane);  // v = 3x3 conv
  conv_row<5>(x, wpk7, b7, Ka, b, h, lane);  // k = 5x5 conv
}

__global__ __launch_bounds__(256) void attn_kernel(
    const float* __restrict__ Qa, const float* __restrict__ Ka,
    const float* __restrict__ Va, float* __restrict__ Xo) {
  __shared__ float smem[8 * 768];  // per wave: K tile(256) V tile(256) P tile(256)
  int bh = blockIdx.x;             // 128 = B*heads
  int b = bh >> 4;
  int head = bh & 15;
  int wave = threadIdx.x >> 5;
  int lane = threadIdx.x & 31;
  int lrow = lane & 15;
  int lhalf = lane >> 4;

  float* Kt = smem + wave * 768;
  float* Vt = Kt + 256;
  float* Pt = Kt + 512;

  const float* Qb = Qa + (size_t)bh * 1024 * 16;
  const float* Kb = Ka + (size_t)bh * 1024 * 16;
  const float* Vb = Va + (size_t)bh * 1024 * 16;

  for (int i = 0; i < 8; ++i) {           // 8 query tiles per wave
    int m0 = (wave * 8 + i) * 16;
    // Q fragments (A layout), d = c*4 + 2*lhalf + j
    v2f qf[4];
#pragma unroll
    for (int c = 0; c < 4; ++c) {
      qf[c] = *(const v2f*)(Qb + (size_t)(m0 + lrow) * 16 + c * 4 + 2 * lhalf);
    }
    v8f O = {};
    float rmax[8], rsum[8];
#pragma unroll
    for (int r = 0; r < 8; ++r) { rmax[r] = -3.4e38f; rsum[r] = 0.0f; }

#pragma unroll 1
    for (int kt = 0; kt < 64; ++kt) {     // stream over key tiles
      // Stage 16x16 K and V tiles into LDS (b128 loads; DS is in-order/wave)
      const float4* ks = (const float4*)(Kb + (size_t)(kt * 16 + lrow) * 16 + lhalf * 8);
      const float4* vs = (const float4*)(Vb + (size_t)(kt * 16 + lrow) * 16 + lhalf * 8);
      float4* kd = (float4*)(Kt + lrow * 16 + lhalf * 8);
      float4* vd = (float4*)(Vt + lrow * 16 + lhalf * 8);
      kd[0] = ks[0]; kd[1] = ks[1];
      vd[0] = vs[0]; vd[1] = vs[1];

      // S = Q * K^T   (B frag: K[m'=lrow][d = c*4 + 2*j + lhalf])
      v8f S = {};
#pragma unroll
      for (int c = 0; c < 4; ++c) {
        v2f kf;
        kf[0] = Kt[lrow * 16 + c * 4 + lhalf];
        kf[1] = Kt[lrow * 16 + c * 4 + lhalf + 2];
        S = wmma_f32(qf[c], kf, S);
      }

      // Online softmax: rows live in 16-lane halves -> butterfly shuffles
#pragma unroll
      for (int r = 0; r < 8; ++r) {
        float s = S[r] * 0.25f;  // scale = 16^-0.5
        float rm = s;
        rm = fmaxf(rm, __shfl_xor(rm, 1, 32));
        rm = fmaxf(rm, __shfl_xor(rm, 2, 32));
        rm = fmaxf(rm, __shfl_xor(rm, 4, 32));
        rm = fmaxf(rm, __shfl_xor(rm, 8, 32));
        float mn = fmaxf(rmax[r], rm);
        float corr = __expf(rmax[r] - mn);
        rmax[r] = mn;
        float p = __expf(s - mn);
        float ps = p;
        ps += __shfl_xor(ps, 1, 32);
        ps += __shfl_xor(ps, 2, 32);
        ps += __shfl_xor(ps, 4, 32);
        ps += __shfl_xor(ps, 8, 32);
        rsum[r] = rsum[r] * corr + ps;
        O[r] *= corr;
        Pt[(r + 8 * lhalf) * 16 + lrow] = p;  // row-major P for A-relayout
      }

      // O += P * V  (A frag from Pt, B frag from Vt)
#pragma unroll
      for (int c = 0; c < 4; ++c) {
        v2f pf, vf;
        pf = *(const v2f*)(Pt + lrow * 16 + c * 4 + 2 * lhalf);
        vf[0] = Vt[(c * 4 + lhalf) * 16 + lrow];
        vf[1] = Vt[(c * 4 + lhalf + 2) * 16 + lrow];
        O = wmma_f32(pf, vf, O);
      }
    }

    // Normalize and store in projection layout: row = token m, col = head*16+d
#pragma unroll
    for (int r = 0; r < 8; ++r) {
      int m = m0 + r + 8 * lhalf;
      Xo[((size_t)b * 1024 + m) * 256 + head * 16 + lrow] = O[r] / rsum[r];
    }
  }
}

__global__ __launch_bounds__(256) void proj_kernel(
    const float* __restrict__ Xo, const float* __restrict__ wppk,
    float* __restrict__ y) {
  int wave = threadIdx.x >> 5;
  int lane = threadIdx.x & 31;
  int job = blockIdx.x * 8 + wave;  // 8192 = (8192/16 row tiles) * 16 col tiles
  int rt = job >> 4;
  int ct = job & 15;
  int lrow = lane & 15;
  int lhalf = lane >> 4;

  const float* ap = Xo + (size_t)(rt * 16 + lrow) * 256 + 2 * lhalf;
  const float* bp = wppk + ((size_t)ct * 32 + lane) * 2;  // packed, tap=0

  v8f acc = {};
#pragma unroll 4
  for (int c = 0; c < 64; ++c) {
    v2f a = *(const v2f*)(ap + c * 4);               // A: coalesced b64
    v2f bf = *(const v2f*)(bp + (size_t)c * 1024);   // B: coalesced b64 (packed)
    acc = wmma_f32(a, bf, acc);
  }
#pragma unroll
  for (int r = 0; r < 8; ++r) {
    y[(size_t)(rt * 16 + r + 8 * lhalf) * 256 + ct * 16 + lrow] = acc[r];
  }
}

extern "C" void kernel_launch(void* const* d_in, const int* in_sizes, int n_in,
                              void* d_out, int out_size, void* d_ws, size_t ws_size,
                              hipStream_t stream) {
  const float* x  = (const float*)d_in[0];
  const float* w3 = (const float*)d_in[1];
  const float* b3 = (const float*)d_in[2];
  const float* w5 = (const float*)d_in[3];
  const float* b5 = (const float*)d_in[4];
  const float* w7 = (const float*)d_in[5];
  const float* b7 = (const float*)d_in[6];
  const float* wp = (const float*)d_in[7];

  // Workspace layout (floats):
  //   Qa | Ka | Va | Xo          : 4 x 2,097,152        (32 MB)
  //   wpk3 | wpk5 | wpk7 | wppk  : (1+9+25+1) x 65,536  (~9.4 MB)
  const size_t SEG = (size_t)8 * 16 * 1024 * 16;  // 2,097,152
  const size_t WSEG = 64 * 16 * 64;               // 65,536 per tap
  float* Qa = (float*)d_ws;
  float* Ka = Qa + SEG;
  float* Va = Ka + SEG;
  float* Xo = Va + SEG;
  float* wpk3 = Xo + SEG;
  float* wpk5 = wpk3 + 1 * WSEG;
  float* wpk7 = wpk5 + 9 * WSEG;
  float* wppk = wpk7 + 25 * WSEG;

  // Stage 0: fragment-order weight repack (one coalesced b64 per B fragment).
  pack_w_kernel<<<(int)((1 * WSEG + 255) / 256), 256, 0, stream>>>(w3, wpk3, 1, (int)(1 * WSEG));
  pack_w_kernel<<<(int)((9 * WSEG + 255) / 256), 256, 0, stream>>>(w5, wpk5, 9, (int)(9 * WSEG));
  pack_w_kernel<<<(int)((25 * WSEG + 255) / 256), 256, 0, stream>>>(w7, wpk7, 25, (int)(25 * WSEG));
  pack_w_kernel<<<(int)((1 * WSEG + 255) / 256), 256, 0, stream>>>(wp, wppk, 1, (int)(1 * WSEG));

  conv_qkv_kernel<<<256, 256, 0, stream>>>(x, wpk3, b3, wpk5, b5, wpk7, b7, Qa, Va, Ka);
  attn_kernel<<<128, 256, 0, stream>>>(Qa, Ka, Va, Xo);
  proj_kernel<<<1024, 256, 0, stream>>>(Xo, wppk, (float*)d_out);
}